// NewSepConv_37898791420124
// MI455X (gfx1250) — compile-verified
//
#include <hip/hip_runtime.h>

typedef float v2f __attribute__((ext_vector_type(2)));
typedef float v8f __attribute__((ext_vector_type(8)));

namespace {
constexpr int KK   = 17;   // taps
constexpr int PAD  = 8;    // replication pad
constexpr int NB   = 4;    // batch
constexpr int NC   = 3;    // channels
constexpr int HH   = 256;
constexpr int WW   = 256;
constexpr int TILE = 16;   // output pixels per wave
}

__device__ __forceinline__ int clampi(int x, int lo, int hi) {
  return x < lo ? lo : (x > hi ? hi : x);
}

// Adaptive separable conv, fully inside the f32 WMMA accumulator:
//   out[c,p] = diag_p( Sum_u Sum_chunk  ToeplitzA(img,row(u),chunk) x B'(u,chunk) )
//   with B'[k,n] = hors[v0+k, n] * vers[u, n]   (column-scaled B folds the
//   vertical weights into the matmul so the diagonal needs extracting only
//   once per channel).
// One wave (32 lanes) per 16-pixel tile; blockDim=32 so all tile/base address
// math is wave-uniform (SALU), lanes carry only matrix fragments + offsets.
__global__ __launch_bounds__(32) void sepconv_wmma_f32(
    const float* __restrict__ imgs, const float* __restrict__ vers,
    const float* __restrict__ hors, float* __restrict__ out)
{
  const int lane = (int)threadIdx.x;       // 0..31
  const int tile = (int)blockIdx.x;        // uniform -> SGPR math

  const int tilesPerRow = WW / TILE;       // 16
  const int b   = tile / (HH * tilesPerRow);
  const int rem = tile - b * HH * tilesPerRow;
  const int h   = rem / tilesPerRow;
  const int w0  = (rem - h * tilesPerRow) * TILE;

  const long plane = (long)HH * WW;

  // Lane roles (wave32):
  //  A 16x4 f32: lane L -> row M=L&15, VGPR0/1 -> K=kbase,kbase+1 (kbase=0|2)
  //  B 4x16 f32: lane L -> col N=L&15, VGPR0/1 -> K=kbase,kbase+1
  const int nm    = lane & 15;
  const int kbase = (lane < 16) ? 0 : 2;

  // ---- clamped Toeplitz column offsets: invariant across (c,u), compute once
  int offA[5][2];
  #pragma unroll
  for (int c5 = 0; c5 < 5; ++c5) {
    const int cx = w0 - PAD + 4 * c5 + nm + kbase;
    offA[c5][0] = clampi(cx,     0, WW - 1);
    offA[c5][1] = clampi(cx + 1, 0, WW - 1);
  }

  // ---- preload raw B chunks from hors (zero-padded past tap 16)
  v2f Bc[5];
  {
    const float* hbase = hors + ((long)b * KK) * plane + (long)h * WW + w0;
    #pragma unroll
    for (int c5 = 0; c5 < 5; ++c5) {
      const int k0 = 4 * c5 + kbase;
      Bc[c5].x = (k0     < KK) ? hbase[(long)k0 * plane + nm]       : 0.0f;
      Bc[c5].y = (k0 + 1 < KK) ? hbase[(long)(k0 + 1) * plane + nm] : 0.0f;
    }
  }

  const float* vbase = vers + ((long)b * KK) * plane + (long)h * WW + w0;
  const float* ibase = imgs + ((long)b * NC) * plane;

  // One f32 accumulator per channel; all 17*5 = 85 WMMAs per channel
  // accumulate into it. Rotating 3 accumulators breaks dependent chains.
  v8f C[NC];
  #pragma unroll
  for (int c = 0; c < NC; ++c) C[c] = (v8f){};

  for (int u = 0; u < KK; ++u) {
    // column-scale B by this tap's vertical weight (shared by all channels)
    const float vu = vbase[(long)u * plane + nm];
    v2f Bp[5];
    #pragma unroll
    for (int c5 = 0; c5 < 5; ++c5) {
      Bp[c5].x = Bc[c5].x * vu;
      Bp[c5].y = Bc[c5].y * vu;
    }

    const long rowoff = (long)clampi(h - PAD + u, 0, HH - 1) * WW;  // replication pad
    #pragma unroll
    for (int c = 0; c < NC; ++c) {
      const float* row = ibase + (long)c * plane + rowoff;  // uniform base (SGPR)
      #pragma unroll
      for (int c5 = 0; c5 < 5; ++c5) {
        v2f A;
        A.x = row[offA[c5][0]];
        A.y = row[offA[c5][1]];
        C[c] = __builtin_amdgcn_wmma_f32_16x16x4_f32(
                   false, A, false, Bp[c5], (short)0, C[c], false, false);
      }
    }
  }

  // ---- diagonal extraction (3x total) + store
  // D[p][p]: lane p (p<8) holds it at C[p]; lane p+16 (p>=8) at C[p-8].
  const bool active = (lane < 8) || (lane >= 24);
  const int  p      = (lane < 8) ? lane : ((lane >= 24) ? (lane - 16) : 0);
  const int  dsel   = lane & 7;
  float* obase = out + ((long)b * NC) * plane + (long)h * WW + w0;

  #pragma unroll
  for (int c = 0; c < NC; ++c) {
    float d = C[c][0];
    d = (dsel == 1) ? C[c][1] : d;
    d = (dsel == 2) ? C[c][2] : d;
    d = (dsel == 3) ? C[c][3] : d;
    d = (dsel == 4) ? C[c][4] : d;
    d = (dsel == 5) ? C[c][5] : d;
    d = (dsel == 6) ? C[c][6] : d;
    d = (dsel == 7) ? C[c][7] : d;
    if (active) obase[(long)c * plane + p] = d;
  }
}

extern "C" void kernel_launch(void* const* d_in, const int* in_sizes, int n_in,
                              void* d_out, int out_size, void* d_ws, size_t ws_size,
                              hipStream_t stream) {
  const float* imgs = (const float*)d_in[0];
  const float* vers = (const float*)d_in[1];
  const float* hors = (const float*)d_in[2];
  float* out = (float*)d_out;

  const int totalTiles = NB * HH * (WW / TILE);   // 16384 tiles, one wave each
  dim3 grid(totalTiles);
  dim3 block(32);                                 // one wave32 per block
  sepconv_wmma_f32<<<grid, block, 0, stream>>>(imgs, vers, hors, out);
}